// MultiHeadAttentionWeights_71476845740210
// MI455X (gfx1250) — compile-verified
//
#include <hip/hip_runtime.h>

typedef __attribute__((ext_vector_type(2))) float v2f;
typedef __attribute__((ext_vector_type(8))) float v8f;

#define EMBED   512
#define NHEADS  8
#define HD      64
#define SEQT    1024
#define NBATCH  8
#define RELROWS 2047
#define KBLK    32
#define KPAD    36   // row pitch: 144B = 16B-aligned rows, conflict-free fragments

union AccU { v8f v; float f[8]; };

__device__ __forceinline__ v8f wmma_f32x4(v2f a, v2f b, v8f c) {
  // D = A(16x4) * B(4x16) + C(16x16), fp32, wave32
  return __builtin_amdgcn_wmma_f32_16x16x4_f32(
      /*neg_a=*/false, a, /*neg_b=*/false, b,
      /*c_mod=*/(short)0, c, /*reuse_a=*/false, /*reuse_b=*/false);
}

// Low 32 bits of a flat pointer to __shared__ = wave-relative LDS byte offset
// (flat LDS addresses are {aperture_hi, lds_offset32}; HW truncates to [31:0]).
__device__ __forceinline__ unsigned lds_off(const void* p) {
  return (unsigned)(unsigned long long)p;
}

__device__ __forceinline__ void async_ld_b128(unsigned lds, const float* g) {
  asm volatile("global_load_async_to_lds_b128 %0, %1, off"
               :: "v"(lds), "v"(g) : "memory");
}
__device__ __forceinline__ void async_ld_b32(unsigned lds, const float* g) {
  asm volatile("global_load_async_to_lds_b32 %0, %1, off"
               :: "v"(lds), "v"(g) : "memory");
}
__device__ __forceinline__ void wait_async0() {
  asm volatile("s_wait_asynccnt 0x0" ::: "memory");
}

// ---------------------------------------------------------------------------
// C[M x 512] = A[M x 512] @ W[512 x 512]  (fp32, WMMA 16x16x4)
// Block: 128 threads = 4 waves; 64x64 output tile per block.
// K is processed in 32-wide blocks, double-buffered: async DMA of block kb+1
// overlaps WMMA on block kb. W is staged transposed so a B fragment is one
// aligned 8B LDS load.
// ---------------------------------------------------------------------------
__global__ __launch_bounds__(128)
void gemm512_kernel(const float* __restrict__ A, const float* __restrict__ W,
                    float* __restrict__ C, int M) {
  __shared__ float As[2][64][KPAD];   // [m][k]
  __shared__ float Wt[2][64][KPAD];   // [n][k] (transposed)

  const int tid  = threadIdx.x;
  const int lane = tid & 31;
  const int wave = tid >> 5;
  const int mblk = blockIdx.x * 64;
  const int nblk = blockIdx.y * 64;

  const int am  = lane & 15;
  const int akb = (lane >> 4) * 2;

  // Pre-zero ragged rows (never targeted by async loads -> no write race).
  if (mblk + 64 > M) {
    int oob0 = (M > mblk) ? (M - mblk) : 0;
    for (int r = oob0 + (tid >> 3); r < 64; r += 16)
      for (int c = (tid & 7); c < KPAD; c += 8) {
        As[0][r][c] = 0.0f;
        As[1][r][c] = 0.0f;
      }
    asm volatile("s_wait_dscnt 0x0" ::: "memory");
  }
  __syncthreads();

  auto stage = [&](int kb, int buf) {
    const int k0 = kb * KBLK;
    // A tile 64x32: b128 async per lane.
#pragma unroll
    for (int t = 0; t < 4; ++t) {
      int idx = tid + t * 128;        // float4 unit, 0..511
      int row = idx >> 3;             // 8 float4 per row
      int c4  = (idx & 7) * 4;
      if (mblk + row < M)
        async_ld_b128(lds_off(&As[buf][row][c4]),
                      A + (size_t)(mblk + row) * EMBED + k0 + c4);
    }
    // W tile 32x64 -> transposed Wt[n][k]: b32 async, global side coalesced.
#pragma unroll
    for (int t = 0; t < 16; ++t) {
      int idx = tid + t * 128;        // 0..2047
      int n = idx & 63;
      int k = idx >> 6;
      async_ld_b32(lds_off(&Wt[buf][n][k]),
                   W + (size_t)(k0 + k) * EMBED + nblk + n);
    }
  };

  AccU acc[4];
#pragma unroll
  for (int cs = 0; cs < 4; ++cs)
#pragma unroll
    for (int i = 0; i < 8; ++i) acc[cs].f[i] = 0.0f;

  stage(0, 0);

  for (int kb = 0; kb < 16; ++kb) {
    const int cur = kb & 1;
    wait_async0();        // this wave's DMA for block kb done
    __syncthreads();      // everyone's DMA landed; prior reads of nxt finished
    if (kb + 1 < 16) stage(kb + 1, cur ^ 1);   // overlap DMA with WMMA

#pragma unroll
    for (int step = 0; step < 8; ++step) {
      const int kk = step * 4;
      float2 av = *(const float2*)&As[cur][wave * 16 + am][kk + akb];
      v2f a; a.x = av.x; a.y = av.y;
#pragma unroll
      for (int cs = 0; cs < 4; ++cs) {
        float2 bv = *(const float2*)&Wt[cur][cs * 16 + am][kk + akb];
        v2f b; b.x = bv.x; b.y = bv.y;
        acc[cs].v = wmma_f32x4(a, b, acc[cs].v);
      }
    }
  }

#pragma unroll
  for (int cs = 0; cs < 4; ++cs)
#pragma unroll
    for (int v = 0; v < 8; ++v) {
      int row = mblk + wave * 16 + v + (lane >> 4) * 8;
      int col = nblk + cs * 16 + am;
      if (row < M) C[(size_t)row * EMBED + col] = acc[cs].f[v];
    }
}

// ---------------------------------------------------------------------------
// Fused relative-position attention weights.
// Grid: (T/16, H, B). Block: 128 threads = 4 waves.
// Wave w owns the s-tiles starting at (w + 4*it)*16, it = 0..15, as 16 register-resident 16x16 score
// tiles: content WMMA + banded pos WMMA (G-window trick, LDS gather) + mask +
// cross-wave softmax + single coalesced write of attn.
// ---------------------------------------------------------------------------
__global__ __launch_bounds__(128)
void relattn_kernel(const float* __restrict__ Q, const float* __restrict__ K,
                    const float* __restrict__ P, const float* __restrict__ REL,
                    const unsigned char* __restrict__ mask,
                    float* __restrict__ out) {
  __shared__ float qs[16][68];
  __shared__ float ps[16][68];
  __shared__ float gbuf[4][16][33];   // per-wave G-tile (16 x 32) scratch
  __shared__ float red[16][65];       // per-row cross-wave partials
  __shared__ float rowmaxs[16];
  __shared__ float rowinv[16];

  const int tid  = threadIdx.x;
  const int lane = tid & 31;
  const int wave = tid >> 5;
  const int t0   = blockIdx.x * 16;
  const int h    = blockIdx.y;
  const int b    = blockIdx.z;

  const int am  = lane & 15;
  const int hi  = lane >> 4;
  const int akb = hi * 2;
  const int gm  = hi * 8;

  const size_t headBase = (size_t)b * SEQT * EMBED + (size_t)h * HD;

  // Async-stage q and p blocks (16 x 64 each) straight into LDS.
#pragma unroll
  for (int t = 0; t < 2; ++t) {
    int idx = tid + t * 128;          // float4 unit, 0..255
    int row = idx >> 4;
    int c4  = (idx & 15) * 4;
    async_ld_b128(lds_off(&qs[row][c4]),
                  Q + headBase + (size_t)(t0 + row) * EMBED + c4);
    async_ld_b128(lds_off(&ps[row][c4]),
                  P + headBase + (size_t)(t0 + row) * EMBED + c4);
  }
  wait_async0();
  __syncthreads();

  // A-fragments for all 16 K-steps (q pre-scaled by dq^-0.5 = 0.125 here,
  // since async DMA cannot scale in flight).
  v2f aq[16], ap[16];
#pragma unroll
  for (int s = 0; s < 16; ++s) {
    aq[s].x = qs[am][s * 4 + akb] * 0.125f;
    aq[s].y = qs[am][s * 4 + akb + 1] * 0.125f;
    ap[s].x = ps[am][s * 4 + akb];
    ap[s].y = ps[am][s * 4 + akb + 1];
  }

  const float* kcol   = K + headBase;           // + s*EMBED + kidx
  const float* relcol = REL + (size_t)h * HD;   // + r*EMBED + kidx

  AccU c[16];   // 16 register-resident 16x16 score tiles per wave

#pragma unroll
  for (int it = 0; it < 16; ++it) {
    const int s0    = (wave + it * 4) * 16;
    const int rbase = t0 - s0 + 1008;           // rel window base (>= 0)

    AccU cc, g0, g1;
#pragma unroll
    for (int i = 0; i < 8; ++i) { cc.f[i] = 0.f; g0.f[i] = 0.f; g1.f[i] = 0.f; }

    const int r0idx = rbase + am;                          // <= 2031
    int r1idx = rbase + 16 + am;                           // could hit 2047
    if (r1idx > RELROWS - 1) r1idx = RELROWS - 1;          // j=31 never gathered

#pragma unroll
    for (int step = 0; step < 16; ++step) {
      const int kidx = step * 4 + akb;
      float2 kv = *(const float2*)(kcol + (size_t)(s0 + am) * EMBED + kidx);
      v2f bk; bk.x = kv.x; bk.y = kv.y;
      cc.v = wmma_f32x4(aq[step], bk, cc.v);

      float2 r0 = *(const float2*)(relcol + (size_t)r0idx * EMBED + kidx);
      v2f b0; b0.x = r0.x; b0.y = r0.y;
      g0.v = wmma_f32x4(ap[step], b0, g0.v);

      float2 r1 = *(const float2*)(relcol + (size_t)r1idx * EMBED + kidx);
      v2f b1; b1.x = r1.x; b1.y = r1.y;
      g1.v = wmma_f32x4(ap[step], b1, g1.v);
    }

    // Dump G (16x32) to this wave's LDS scratch, then gather the band
    // j = ti - si + 15 and fold into the content tile.
#pragma unroll
    for (int v = 0; v < 8; ++v) {
      gbuf[wave][gm + v][am]      = g0.f[v];
      gbuf[wave][gm + v][16 + am] = g1.f[v];
    }
    asm volatile("s_wait_dscnt 0x0" ::: "memory");   // per-wave LDS RAW order

#pragma unroll
    for (int v = 0; v < 8; ++v) {
      const int ti = gm + v;
      const int si = am;
      const int j  = ti - si + 15;                 // 0..30
      float val = cc.f[v] + gbuf[wave][ti][j];
      if (!mask[(size_t)b * SEQT + s0 + si]) val = -10000.0f;
      c[it].f[v] = val;
    }
  }

  // ---- softmax over s (cross-wave via LDS partials) ----
  float pmax[8];
#pragma unroll
  for (int v = 0; v < 8; ++v) {
    float m = -3.0e38f;
#pragma unroll
    for (int it = 0; it < 16; ++it) m = fmaxf(m, c[it].f[v]);
    pmax[v] = m;
  }
#pragma unroll
  for (int v = 0; v < 8; ++v) red[gm + v][wave * 16 + am] = pmax[v];
  __syncthreads();
  if (tid < 16) {
    float m = red[tid][0];
    for (int i = 1; i < 64; ++i) m = fmaxf(m, red[tid][i]);
    rowmaxs[tid] = m;
  }
  __syncthreads();

  float psum[8];
#pragma unroll
  for (int v = 0; v < 8; ++v) psum[v] = 0.0f;
#pragma unroll
  for (int it = 0; it < 16; ++it)
#pragma unroll
    for (int v = 0; v < 8; ++v) {
      float e = __expf(c[it].f[v] - rowmaxs[gm + v]);
      c[it].f[v] = e;
      psum[v] += e;
    }
#pragma unroll
  for (int v = 0; v < 8; ++v) red[gm + v][wave * 16 + am] = psum[v];
  __syncthreads();
  if (tid < 16) {
    float s = 0.0f;
    for (int i = 0; i < 64; ++i) s += red[tid][i];
    rowinv[tid] = (s > 0.0f) ? (1.0f / s) : 0.0f;
  }
  __syncthreads();

  // ---- normalize, apply t-mask, single write of attn (write-BW bound) ----
  const size_t outbase = ((size_t)(b * NHEADS + h) * SEQT + t0) * SEQT;
#pragma unroll
  for (int it = 0; it < 16; ++it) {
    const int s0 = (wave + it * 4) * 16;
#pragma unroll
    for (int v = 0; v < 8; ++v) {
      const int ti = gm + v;
      float val = c[it].f[v] * rowinv[ti];
      if (!mask[(size_t)b * SEQT + t0 + ti]) val = 0.0f;
      out[outbase + (size_t)ti * SEQT + s0 + am] = val;
    }
  }
}

// ---------------------------------------------------------------------------
extern "C" void kernel_launch(void* const* d_in, const int* in_sizes, int n_in,
                              void* d_out, int out_size, void* d_ws, size_t ws_size,
                              hipStream_t stream) {
  (void)in_sizes; (void)n_in; (void)out_size; (void)ws_size;
  const float* x        = (const float*)d_in[0];
  const float* pos_emb  = (const float*)d_in[1];
  const unsigned char* mask = (const unsigned char*)d_in[2];
  const float* Wq   = (const float*)d_in[3];
  const float* Wk   = (const float*)d_in[4];
  const float* Wpq  = (const float*)d_in[5];
  const float* Wpos = (const float*)d_in[6];

  float* qws   = (float*)d_ws;                       // [8192 x 512]
  float* kws   = qws + (size_t)NBATCH * SEQT * EMBED;
  float* pws   = kws + (size_t)NBATCH * SEQT * EMBED;
  float* relws = pws + (size_t)NBATCH * SEQT * EMBED; // [2047 x 512]

  dim3 blk(128);
  gemm512_kernel<<<dim3(128, 8), blk, 0, stream>>>(x, Wq, qws, NBATCH * SEQT);
  gemm512_kernel<<<dim3(128, 8), blk, 0, stream>>>(x, Wk, kws, NBATCH * SEQT);
  gemm512_kernel<<<dim3(128, 8), blk, 0, stream>>>(x, Wpq, pws, NBATCH * SEQT);
  gemm512_kernel<<<dim3(32, 8),  blk, 0, stream>>>(pos_emb, Wpos, relws, RELROWS);

  relattn_kernel<<<dim3(SEQT / 16, NHEADS, NBATCH), blk, 0, stream>>>(
      qws, kws, pws, relws, mask, (float*)d_out);
}